// PeriodCapsuleAggregator_55293408968718
// MI455X (gfx1250) — compile-verified
//
#include <hip/hip_runtime.h>
#include <math.h>

// ---- problem constants (from reference) ----
#define TOKENS   65536      // B*L
#define C_DIM    128
#define K_CAPS   6
#define PDP      48         // P*Dp
#define DDQ      48         // D*Dd
#define D_CAPS   4
#define DD       12         // Dd
#define ITERS    2
#define EPSF     1e-9f

typedef float v2f  __attribute__((ext_vector_type(2)));
typedef float v8f  __attribute__((ext_vector_type(8)));
typedef float f4v  __attribute__((ext_vector_type(4)));
typedef int   i4v  __attribute__((ext_vector_type(4)));

typedef __attribute__((address_space(1))) i4v* glob_i4p;   // global (device) int4*
typedef __attribute__((address_space(3))) i4v* lds_i4p;    // LDS (shared) int4*

// =====================================================================
// Kernel 1: Xp = squash( X_rows[M=B*L*K, C=128] @ Wp^T + bp )
//   - block = 96 threads (3 waves), handles 48 rows = 8 tokens
//   - X token block [128,6] is contiguous; stage into LDS via async
//     global->LDS DMA (ASYNCcnt-tracked) to resolve the stride-6
//     (K-innermost) access pattern for A-fragments
//   - WMMA f32 16x16x4, 32 K-steps, N = 48 = 3 tiles of 16
// =====================================================================
__global__ __launch_bounds__(96)
void proj_squash_kernel(const float* __restrict__ X,
                        const float* __restrict__ Wp,
                        const float* __restrict__ bp,
                        float* __restrict__ Xp)
{
    __shared__ float ldsX[8 * C_DIM * K_CAPS];       // 6144 floats = 24 KB

    const int tid  = threadIdx.x;
    const int lane = tid & 31;
    const int w    = tid >> 5;       // wave 0..2
    const int lm   = lane & 15;
    const int hi   = lane >> 4;

    // ---- stage 8 tokens of X (1536 x b128), coalesced ----
    const f4v* Xg4 = (const f4v*)(X + (size_t)blockIdx.x * (8 * C_DIM * K_CAPS));
    f4v* L4 = (f4v*)ldsX;
#if __has_builtin(__builtin_amdgcn_global_load_async_to_lds_b128)
    // Async DMA straight into LDS (ASYNCcnt-tracked); no VGPR round trip.
    #pragma unroll
    for (int i = 0; i < 16; ++i) {
        __builtin_amdgcn_global_load_async_to_lds_b128(
            (glob_i4p)(void*)(Xg4 + i * 96 + tid),
            (lds_i4p)(void*)(L4 + i * 96 + tid),
            0, 0);
    }
  #if __has_builtin(__builtin_amdgcn_s_wait_asynccnt)
    __builtin_amdgcn_s_wait_asynccnt(0);
  #else
    asm volatile("s_wait_asynccnt 0" ::: "memory");
  #endif
#else
    #pragma unroll
    for (int i = 0; i < 16; ++i)
        L4[i * 96 + tid] = Xg4[i * 96 + tid];
#endif
    __syncthreads();

    // A-fragment source row for this lane: local row r = w*16 + lm
    const int r   = w * 16 + lm;          // 0..47
    const int tok = r / K_CAPS;           // 0..7
    const int kk  = r - tok * K_CAPS;     // 0..5
    const float* arow = ldsX + tok * (C_DIM * K_CAPS) + kk;   // stride-6 over c

    v8f acc[3] = {v8f{}, v8f{}, v8f{}};

    #pragma unroll
    for (int s = 0; s < 32; ++s) {
        const int c = 4 * s + 2 * hi;
        v2f a;
        a.x = arow[c * K_CAPS];
        a.y = arow[(c + 1) * K_CAPS];
        #pragma unroll
        for (int nt = 0; nt < 3; ++nt) {
            const int n = nt * 16 + lm;
            v2f b = *(const v2f*)(Wp + n * C_DIM + c);    // 8B aligned (c even)
            acc[nt] = __builtin_amdgcn_wmma_f32_16x16x4_f32(
                false, a, false, b, (short)0, acc[nt], false, false);
        }
    }

    // ---- epilogue: bias + squash(groups of 8 cols) + store ----
    // C/D layout: VGPR r8 -> row (r8 + 8*hi), lane%16 -> column.
    // A squash group = 8 consecutive columns = 8 consecutive lanes in a
    // 16-lane half; xor masks 1/2/4 stay inside each 8-lane group.
    const long rowBase = (long)blockIdx.x * 48 + w * 16;
    #pragma unroll
    for (int nt = 0; nt < 3; ++nt) {
        const int n = nt * 16 + lm;
        const float bias = bp[n];
        #pragma unroll
        for (int r8 = 0; r8 < 8; ++r8) {
            float x  = acc[nt][r8] + bias;
            float s2 = x * x;
            s2 += __shfl_xor(s2, 1, 32);
            s2 += __shfl_xor(s2, 2, 32);
            s2 += __shfl_xor(s2, 4, 32);
            const float scale = s2 / (1.0f + s2) / (sqrtf(s2) + EPSF);
            const long row = rowBase + r8 + 8 * hi;
            Xp[row * PDP + n] = x * scale;
        }
    }
}

// =====================================================================
// Kernel 2: fused u_hat GEMM + dynamic routing
//   - block = 192 threads (6 waves), 16 tokens per block
//   - wave k computes u_hat[k] tile [16 tokens x 48] via WMMA -> LDS
//   - threads 0..63 (token x d) run 2 routing iterations; softmax over
//     D=4 via shuffles in aligned 4-lane groups; write v to d_out
// =====================================================================
__global__ __launch_bounds__(192)
void uhat_route_kernel(const float* __restrict__ Xp,
                       const float* __restrict__ Wg,   // [K][D*Dd][P*Dp]
                       float* __restrict__ out)
{
    __shared__ float ldsU[K_CAPS * 16 * DDQ];          // 4608 floats = 18 KB

    const int tid  = threadIdx.x;
    const int lane = tid & 31;
    const int w    = tid >> 5;       // wave id == k, 0..5
    const int lm   = lane & 15;
    const int hi   = lane >> 4;
    const int t0   = blockIdx.x * 16;

    // ---- GEMM: u_hat[k=w][token=0..15][n=0..47] ----
    const float* arow = Xp + ((size_t)(t0 + lm) * K_CAPS + w) * PDP;
    const float* wk   = Wg + (size_t)w * DDQ * PDP;

    v8f acc[3] = {v8f{}, v8f{}, v8f{}};
    #pragma unroll
    for (int s = 0; s < 12; ++s) {
        const int p = 4 * s + 2 * hi;
        v2f a = *(const v2f*)(arow + p);               // 8B aligned
        #pragma unroll
        for (int nt = 0; nt < 3; ++nt) {
            const int n = nt * 16 + lm;
            v2f b = *(const v2f*)(wk + n * PDP + p);   // 8B aligned
            acc[nt] = __builtin_amdgcn_wmma_f32_16x16x4_f32(
                false, a, false, b, (short)0, acc[nt], false, false);
        }
    }
    #pragma unroll
    for (int nt = 0; nt < 3; ++nt) {
        #pragma unroll
        for (int r8 = 0; r8 < 8; ++r8) {
            const int trow = r8 + 8 * hi;              // token 0..15
            ldsU[(w * 16 + trow) * DDQ + nt * 16 + lm] = acc[nt][r8];
        }
    }
    __syncthreads();

    // ---- routing: one thread per (token, d); full first two waves ----
    if (tid < 64) {
        const int tl = tid >> 2;     // token 0..15
        const int d  = tid & 3;      // capsule d 0..3

        float u[K_CAPS][DD];
        #pragma unroll
        for (int k = 0; k < K_CAPS; ++k)
            #pragma unroll
            for (int q = 0; q < DD; ++q)
                u[k][q] = ldsU[(k * 16 + tl) * DDQ + d * DD + q];

        float bb[K_CAPS] = {0.f, 0.f, 0.f, 0.f, 0.f, 0.f};
        float v[DD];
        #pragma unroll
        for (int it = 0; it < ITERS; ++it) {
            // softmax over D (4 adjacent lanes share a token)
            float cc[K_CAPS];
            #pragma unroll
            for (int k = 0; k < K_CAPS; ++k) {
                float m = bb[k];
                m = fmaxf(m, __shfl_xor(m, 1, 32));
                m = fmaxf(m, __shfl_xor(m, 2, 32));
                float e = __expf(bb[k] - m);
                float sum = e;
                sum += __shfl_xor(sum, 1, 32);
                sum += __shfl_xor(sum, 2, 32);
                cc[k] = e / sum;
            }
            // s = sum_k c[k]*u_hat[k], squash over q
            float s2 = 0.f;
            #pragma unroll
            for (int q = 0; q < DD; ++q) {
                float s = 0.f;
                #pragma unroll
                for (int k = 0; k < K_CAPS; ++k) s += cc[k] * u[k][q];
                v[q] = s;
                s2 += s * s;
            }
            const float scale = s2 / (1.0f + s2) / (sqrtf(s2) + EPSF);
            #pragma unroll
            for (int q = 0; q < DD; ++q) v[q] *= scale;
            // b update
            #pragma unroll
            for (int k = 0; k < K_CAPS; ++k) {
                float dot = 0.f;
                #pragma unroll
                for (int q = 0; q < DD; ++q) dot += u[k][q] * v[q];
                bb[k] += dot;
            }
        }
        float* o = out + (size_t)(t0 + tl) * DDQ + d * DD;
        #pragma unroll
        for (int q = 0; q < DD; ++q) o[q] = v[q];
    }
}

// =====================================================================
extern "C" void kernel_launch(void* const* d_in, const int* in_sizes, int n_in,
                              void* d_out, int out_size, void* d_ws, size_t ws_size,
                              hipStream_t stream)
{
    (void)in_sizes; (void)n_in; (void)out_size; (void)ws_size;

    const float* X  = (const float*)d_in[0];   // [B,L,C,K]
    const float* Wp = (const float*)d_in[1];   // [P*Dp, C]
    const float* bp = (const float*)d_in[2];   // [P*Dp]
    const float* W  = (const float*)d_in[3];   // [K, D, Dd, P*Dp]

    float* Xp = (float*)d_ws;                  // [B*L*K, 48] = 75.5 MB

    // 393216 rows / 48 rows per block
    proj_squash_kernel<<<dim3(TOKENS * K_CAPS / 48), dim3(96), 0, stream>>>(
        X, Wp, bp, Xp);

    // 65536 tokens / 16 tokens per block
    uhat_route_kernel<<<dim3(TOKENS / 16), dim3(192), 0, stream>>>(
        Xp, W, (float*)d_out);
}